// KNNLoss_18554258719494
// MI455X (gfx1250) — compile-verified
//
#include <hip/hip_runtime.h>
#include <math.h>
#include <stdint.h>

typedef float v2f __attribute__((ext_vector_type(2)));
typedef float v4f __attribute__((ext_vector_type(4)));
typedef float v8f __attribute__((ext_vector_type(8)));

#define DIM 64
#define TILE 16
#define WAVES_PER_BLOCK 8
#define BLOCK_THREADS 256
#define LDS_STRIDE 68   // 16 rows x 64 cols padded: bank = (4n + k) % 64, conflict-free

// ---- CDNA5 async global->LDS copy (ASYNCcnt-tracked), per-lane B128 ----
__device__ __forceinline__ void async_copy_b128(unsigned lds_byte_off,
                                                const void* gptr) {
  unsigned long long ga = (unsigned long long)gptr;
  asm volatile("global_load_async_to_lds_b128 %0, %1, off"
               :: "v"(lds_byte_off), "v"(ga) : "memory");
}
__device__ __forceinline__ void wait_async0() {
  asm volatile("s_wait_asynccnt 0x0" ::: "memory");
}

// ---------------- kernel 0: squared norms ----------------
__global__ void knn_sqn_kernel(const float* __restrict__ x,
                               float* __restrict__ sqn, int n) {
  int i = blockIdx.x * blockDim.x + threadIdx.x;
  if (i >= n) return;
  const v4f* row = (const v4f*)(x + (size_t)i * DIM);
  float s = 0.f;
#pragma unroll
  for (int q = 0; q < DIM / 4; ++q) {
    v4f v = row[q];
    s += v.x * v.x + v.y * v.y + v.z * v.z + v.w * v.w;
  }
  sqn[i] = s;
}

// ---------------- kernel 1: fused WMMA distance tiles + per-row stats ---------
// One wave32 per 16-row block; A fragments live in VGPRs for the whole pass.
// Column tiles double-buffered in LDS via async global->LDS copies.
__global__ void __launch_bounds__(BLOCK_THREADS)
knn_rows_kernel(const float* __restrict__ x, const float* __restrict__ sqn,
                float* __restrict__ per_point, int n) {
  __shared__ float lds[2][TILE * LDS_STRIDE];

  const int lane = threadIdx.x & 31;
  const int wave = threadIdx.x >> 5;
  const int nl   = lane & 15;   // N index within tile (also M for A-frag loads)
  const int half = lane >> 4;   // lane half: selects K pair / M offset +8

  // Force the per-wave tile index into an SGPR so the diagonal check below
  // becomes a true scalar branch (s_cmp/s_cbranch), not an EXEC-mask diamond.
  const int myblock =
      __builtin_amdgcn_readfirstlane(blockIdx.x * WAVES_PER_BLOCK + wave);
  const int row0 = myblock * TILE;

  // A fragment: 16x4 f32 per step kk (K = 4*kk .. 4*kk+3).
  // ISA layout: lane m = nl, VGPR0 = K=2*half, VGPR1 = K=2*half+1.
  v2f afrag[DIM / 4];
#pragma unroll
  for (int kk = 0; kk < DIM / 4; ++kk) {
    const float* p = x + (size_t)(row0 + nl) * DIM + kk * 4 + 2 * half;
    afrag[kk] = *(const v2f*)p;
  }

  float si[8];
#pragma unroll
  for (int r = 0; r < 8; ++r) si[r] = sqn[row0 + r + 8 * half];

  float denom[8], min1[8], min2[8];
#pragma unroll
  for (int r = 0; r < 8; ++r) {
    denom[r] = 0.f; min1[r] = INFINITY; min2[r] = INFINITY;
  }

  // Per-thread slice for tile staging: 256 threads x 4 floats (one B128 each)
  const int sl_row = (threadIdx.x * 4) >> 6;   // 0..15
  const int sl_col = (threadIdx.x * 4) & 63;   // 0..60 step 4

  const int ntiles = n / TILE;

  // prologue: stage tile 0 into buffer 0
  {
    const float* gp = x + (size_t)(0 * TILE + sl_row) * DIM + sl_col;
    unsigned loff = (unsigned)(uintptr_t)&lds[0][sl_row * LDS_STRIDE + sl_col];
    async_copy_b128(loff, gp);
    wait_async0();
  }
  __syncthreads();

  for (int jt = 0; jt < ntiles; ++jt) {
    const int cur = jt & 1;

    // prefetch next column tile into the other buffer (overlaps compute)
    if (jt + 1 < ntiles) {
      const float* gp = x + (size_t)((jt + 1) * TILE + sl_row) * DIM + sl_col;
      unsigned loff =
          (unsigned)(uintptr_t)&lds[cur ^ 1][sl_row * LDS_STRIDE + sl_col];
      async_copy_b128(loff, gp);
    }

    // Gram tile: D = A(16x64) * B(64x16), 16 x V_WMMA_F32_16X16X4_F32
    v8f acc = {};
#pragma unroll
    for (int kk = 0; kk < DIM / 4; ++kk) {
      v2f b = *(const v2f*)&lds[cur][nl * LDS_STRIDE + kk * 4 + 2 * half];
      acc = __builtin_amdgcn_wmma_f32_16x16x4_f32(false, afrag[kk], false, b,
                                                  (short)0, acc, false, false);
    }

    const float sj = sqn[jt * TILE + nl];

    if (jt != myblock) {   // scalar (SGPR) condition -> s_cbranch, no EXEC churn
      // ---- fast path (511/512 tiles): no diagonal, no per-lane compares ----
#pragma unroll
      for (int r = 0; r < 8; ++r) {
        float sq   = si[r] + sj - 2.0f * acc[r];
        float dist = __builtin_amdgcn_sqrtf(fmaxf(sq, 0.0f));
        denom[r]  += __expf(-dist);
        // branchless running top-2 minimum (invariant min1 <= min2):
        // new min2 = median(dist, min1, min2); new min1 = min(min1, dist)
        min2[r] = __builtin_amdgcn_fmed3f(dist, min1[r], min2[r]);
        min1[r] = fminf(min1[r], dist);
      }
    } else {
      // ---- rare path (1/512 tiles): contains the diagonal element ----
      const int gn = jt * TILE + nl;
#pragma unroll
      for (int r = 0; r < 8; ++r) {
        int gm = row0 + r + 8 * half;
        float sq   = si[r] + sj - 2.0f * acc[r];
        float dist = (gm == gn) ? 0.0f
                                : __builtin_amdgcn_sqrtf(fmaxf(sq, 0.0f));
        denom[r]  += __expf(-dist);               // diag adds exp(0)=1, removed later
        float dtop = (gm == gn) ? INFINITY : dist; // exclude self from top-k
        min2[r] = __builtin_amdgcn_fmed3f(dtop, min1[r], min2[r]);
        min1[r] = fminf(min1[r], dtop);
      }
    }

    wait_async0();     // my slice of the next tile has landed in LDS
    __syncthreads();   // everyone's slices have landed; current buffer free
  }

  // reduce (sum, top-2-min) across the 16 lanes of each half-group (wave32)
#pragma unroll
  for (int r = 0; r < 8; ++r) {
#pragma unroll
    for (int mask = 1; mask < 16; mask <<= 1) {
      denom[r] += __shfl_xor(denom[r], mask, 32);
      float o1 = __shfl_xor(min1[r], mask, 32);
      float o2 = __shfl_xor(min2[r], mask, 32);
      // merge two (min1,min2) pairs:
      // new min2 = min( median(o1, min1, min2), o2 ); new min1 = min(min1, o1)
      min2[r] = fminf(__builtin_amdgcn_fmed3f(o1, min1[r], min2[r]), o2);
      min1[r] = fminf(min1[r], o1);
    }
  }

  if (nl == 0) {
#pragma unroll
    for (int r = 0; r < 8; ++r) {
      int gm = row0 + r + 8 * half;
      // loss_i = mean_k d_nbr + log(denom)  with denom = sum_j exp(-d) - 1
      per_point[gm] = 0.5f * (min1[r] + min2[r]) + __logf(denom[r] - 1.0f);
    }
  }
}

// ---------------- kernel 2: final mean ----------------
__global__ void knn_reduce_kernel(const float* __restrict__ pp,
                                  float* __restrict__ out, int n) {
  __shared__ float sbuf[BLOCK_THREADS];
  float s = 0.f;
  for (int i = threadIdx.x; i < n; i += BLOCK_THREADS) s += pp[i];
  sbuf[threadIdx.x] = s;
  __syncthreads();
  for (int step = BLOCK_THREADS / 2; step > 0; step >>= 1) {
    if ((int)threadIdx.x < step) sbuf[threadIdx.x] += sbuf[threadIdx.x + step];
    __syncthreads();
  }
  if (threadIdx.x == 0) out[0] = sbuf[0] / (float)n;
}

extern "C" void kernel_launch(void* const* d_in, const int* in_sizes, int n_in,
                              void* d_out, int out_size, void* d_ws, size_t ws_size,
                              hipStream_t stream) {
  const float* x = (const float*)d_in[0];
  float* out = (float*)d_out;
  const int n = in_sizes[0] / DIM;           // 8192

  float* sqn = (float*)d_ws;                 // n floats
  float* pp  = sqn + n;                      // n floats

  knn_sqn_kernel<<<(n + 255) / 256, 256, 0, stream>>>(x, sqn, n);

  const int rowblocks = n / (TILE * WAVES_PER_BLOCK);   // 64
  knn_rows_kernel<<<rowblocks, BLOCK_THREADS, 0, stream>>>(x, sqn, pp, n);

  knn_reduce_kernel<<<1, BLOCK_THREADS, 0, stream>>>(pp, out, n);
}